// ConstOutputFilteredNormalized_30528627540481
// MI455X (gfx1250) — compile-verified
//
#include <hip/hip_runtime.h>
#include <stdint.h>

typedef float v4f __attribute__((ext_vector_type(4)));

constexpr int kN   = 4096;              // row length (fixed by problem)
constexpr int kT   = 256;               // threads per block = 8 wave32
constexpr int kVec = 4;                 // floats per b128
constexpr int kLd  = kN / (kT * kVec);  // 4 async b128 loads per thread per row
constexpr int kRpb = 8;                 // rows per block

// Issue one 16KB tile (a whole row) as 4 async b128 global->LDS loads per thread.
// GVS addressing: mem = SGPR64 base + VGPR32 offset; LDS dest = VGPR lds addr.
__device__ __forceinline__ void issue_async_tile(uint32_t lds_base,
                                                 const float* g, int tid) {
  uint64_t base = (uint64_t)(uintptr_t)g;
#pragma unroll
  for (int i = 0; i < kLd; ++i) {
    uint32_t goff  = (uint32_t)((i * kT + tid) * (kVec * 4));
    uint32_t laddr = lds_base + goff;
    asm volatile("global_load_async_to_lds_b128 %0, %1, %2"
                 :
                 : "v"(laddr), "v"(goff), "s"(base)
                 : "memory");
  }
}

__global__ __launch_bounds__(kT) void masked_norm_kernel(
    const float* __restrict__ x, const float* __restrict__ f,
    float* __restrict__ y, int B) {
  __shared__ __align__(16) float sF[kN];        // 16 KB, loaded once per block
  __shared__ __align__(16) float sX[2][kN];     // 2 x 16 KB double buffer
  __shared__ float sRed[kT / 32];               // one partial per wave

  const int tid    = threadIdx.x;
  const int rowBeg = blockIdx.x * kRpb;
  const int rowEnd = (rowBeg + kRpb < B) ? (rowBeg + kRpb) : B;
  if (rowBeg >= B) return;  // uniform per block

  // Low 32 bits of a generic pointer into the LDS aperture = LDS byte offset
  // (aperture base occupies only addr[63:32]).
  uint32_t ldsF  = (uint32_t)(uintptr_t)&sF[0];
  uint32_t ldsX0 = (uint32_t)(uintptr_t)&sX[0][0];
  uint32_t ldsX1 = (uint32_t)(uintptr_t)&sX[1][0];

  // Prologue: stage f and row 0 via the async data path.
  issue_async_tile(ldsF, f, tid);
  issue_async_tile(ldsX0, x + (size_t)rowBeg * kN, tid);
  asm volatile("s_wait_asynccnt 0x0" ::: "memory");
  __syncthreads();

  for (int r = rowBeg; r < rowEnd; ++r) {
    const int buf = (r - rowBeg) & 1;

    // Prefetch next row into the other buffer, keep it in flight while we
    // compute: async loads complete in order, so waiting ASYNCcnt<=4 leaves
    // only the 4 just-issued loads outstanding.
    if (r + 1 < rowEnd) {
      issue_async_tile(buf ? ldsX0 : ldsX1, x + (size_t)(r + 1) * kN, tid);
      asm volatile("s_wait_asynccnt 0x4" ::: "memory");
    } else {
      asm volatile("s_wait_asynccnt 0x0" ::: "memory");
    }
    __syncthreads();  // sX[buf] visible to all waves; also fences sRed reuse

    // Masked values + per-thread partial sum (16 elements/thread).
    const v4f* xv = (const v4f*)&sX[buf][0];
    const v4f* fv = (const v4f*)&sF[0];
    v4f   m[kLd];
    float partial = 0.0f;
#pragma unroll
    for (int i = 0; i < kLd; ++i) {
      const int idx = i * kT + tid;
      v4f xx = xv[idx];
      v4f ff = fv[idx];
      v4f mm;
#pragma unroll
      for (int j = 0; j < 4; ++j) {
        float mv = (xx[j] != 0.0f) ? ff[j] : 0.0f;
        mm[j] = mv;
        partial += mv;
      }
      m[i] = mm;
    }

    // wave32 butterfly reduction, then cross-wave combine in LDS.
#pragma unroll
    for (int off = 16; off > 0; off >>= 1)
      partial += __shfl_xor(partial, off, 32);
    if ((tid & 31) == 0) sRed[tid >> 5] = partial;
    __syncthreads();  // all sX[buf] reads are before this; prefetch into it
                      // only happens after every wave passes this barrier
    float denom = 0.0f;
#pragma unroll
    for (int w = 0; w < kT / 32; ++w) denom += sRed[w];
    const float inv = (denom == 0.0f) ? 1.0f : (1.0f / denom);

    // Streaming non-temporal 128-bit stores: y is written once, never reused.
    float* yrow = y + (size_t)r * kN;
#pragma unroll
    for (int i = 0; i < kLd; ++i) {
      const int idx = i * kT + tid;
      __builtin_nontemporal_store(m[i] * inv, (v4f*)yrow + idx);
    }
  }
}

extern "C" void kernel_launch(void* const* d_in, const int* in_sizes, int n_in,
                              void* d_out, int out_size, void* d_ws, size_t ws_size,
                              hipStream_t stream) {
  (void)n_in; (void)out_size; (void)d_ws; (void)ws_size;
  const float* x = (const float*)d_in[0];
  const float* f = (const float*)d_in[1];
  float*       y = (float*)d_out;

  const int N = in_sizes[1];          // f has N elements
  const int B = in_sizes[0] / N;      // x is B x N
  if (N != kN) return;                // problem size is fixed at 16384 x 4096

  const int grid = (B + kRpb - 1) / kRpb;  // 2048 blocks
  hipLaunchKernelGGL(masked_norm_kernel, dim3(grid), dim3(kT), 0, stream,
                     x, f, y, B);
}